// HybridBlock_27376121544955
// MI455X (gfx1250) — compile-verified
//
#include <hip/hip_runtime.h>
#include <cstddef>

// ---------------------------------------------------------------------------
// Model constants (match the reference)
// ---------------------------------------------------------------------------
#define BB       4
#define LL       2048
#define DD       256
#define D_INNER  512
#define D_STATE  16
#define DT_RANK  16
#define D_CONV   4
#define N_HEADS  4
#define HD       64          // head dim = D / N_HEADS
#define H_MLP    1024
#define ROWS     (BB*LL)     // 8192

typedef float v2f __attribute__((ext_vector_type(2)));
typedef float v8f __attribute__((ext_vector_type(8)));

__device__ __forceinline__ float silu_f(float x) {
    return x / (1.0f + __expf(-x));
}

// ===========================================================================
// LayerNorm: one wave per row of 256 elements (8 per lane)
// ===========================================================================
__global__ void ln_kernel(const float* __restrict__ x,
                          const float* __restrict__ w,
                          const float* __restrict__ b,
                          float* __restrict__ out, int rows)
{
    int wave = threadIdx.x >> 5;
    int lane = threadIdx.x & 31;
    int row  = blockIdx.x * 4 + wave;
    if (row >= rows) return;
    const float* xr = x + (size_t)row * DD;
    float v[8];
    float s = 0.f;
#pragma unroll
    for (int i = 0; i < 8; ++i) { v[i] = xr[lane + i * 32]; s += v[i]; }
#pragma unroll
    for (int off = 16; off >= 1; off >>= 1) s += __shfl_xor(s, off, 32);
    float mu = s * (1.0f / DD);
    float q = 0.f;
#pragma unroll
    for (int i = 0; i < 8; ++i) { float d = v[i] - mu; q += d * d; }
#pragma unroll
    for (int off = 16; off >= 1; off >>= 1) q += __shfl_xor(q, off, 32);
    float inv = rsqrtf(q * (1.0f / DD) + 1e-5f);
    float* orow = out + (size_t)row * DD;
#pragma unroll
    for (int i = 0; i < 8; ++i) {
        int c = lane + i * 32;
        orow[c] = (v[i] - mu) * inv * w[c] + b[c];
    }
}

// ===========================================================================
// Flip along L:  out[b, L-1-l, :] = in[b, l, :]
// ===========================================================================
__global__ void flip_kernel(const float* __restrict__ in, float* __restrict__ out)
{
    size_t idx = (size_t)blockIdx.x * blockDim.x + threadIdx.x;
    if (idx >= (size_t)ROWS * DD) return;
    int d = idx % DD;
    int l = (idx / DD) % LL;
    int b = idx / ((size_t)DD * LL);
    out[((size_t)b * LL + (LL - 1 - l)) * DD + d] = in[idx];
}

// x1 = x + fwd + flip(bwd)
__global__ void combine_kernel(const float* __restrict__ x,
                               const float* __restrict__ fwd,
                               const float* __restrict__ bwd,
                               float* __restrict__ out)
{
    size_t idx = (size_t)blockIdx.x * blockDim.x + threadIdx.x;
    if (idx >= (size_t)ROWS * DD) return;
    int d = idx % DD;
    int l = (idx / DD) % LL;
    int b = idx / ((size_t)DD * LL);
    float bw = bwd[((size_t)b * LL + (LL - 1 - l)) * DD + d];
    out[idx] = x[idx] + fwd[idx] + bw;
}

// ===========================================================================
// Depthwise causal conv (width 4) + bias + SiLU.
// xi lives in xz[:, 0:512] (row stride 1024). Output xc (rows,512).
// ===========================================================================
__global__ void conv_silu_kernel(const float* __restrict__ xz,
                                 const float* __restrict__ convw,
                                 const float* __restrict__ convb,
                                 float* __restrict__ xc)
{
    size_t idx = (size_t)blockIdx.x * blockDim.x + threadIdx.x;
    if (idx >= (size_t)ROWS * D_INNER) return;
    int d = idx % D_INNER;
    int l = (idx / D_INNER) % LL;
    int b = idx / ((size_t)D_INNER * LL);
    float acc = convb[d];
#pragma unroll
    for (int t = 0; t < D_CONV; ++t) {
        int ls = l - (D_CONV - 1) + t;
        if (ls >= 0) {
            float xv = xz[((size_t)b * LL + ls) * (2 * D_INNER) + d];
            acc += convw[d * D_CONV + t] * xv;
        }
    }
    xc[idx] = silu_f(acc);
}

// ===========================================================================
// Selective scan. One thread per (b, d_inner) channel; 16-state in registers.
// dt:(rows,512)  dbl:(rows,48) [B at 16..31, C at 32..47]  u=xc:(rows,512)
// ===========================================================================
__global__ void scan_kernel(const float* __restrict__ dt,
                            const float* __restrict__ dbl,
                            const float* __restrict__ xc,
                            const float* __restrict__ Alog,
                            const float* __restrict__ Dp,
                            float* __restrict__ ycore)
{
    int b = blockIdx.x;
    int d = blockIdx.y * blockDim.x + threadIdx.x;   // 0..511
    float A[D_STATE];
#pragma unroll
    for (int n = 0; n < D_STATE; ++n) A[n] = -__expf(Alog[d * D_STATE + n]);
    float Dpv = Dp[d];
    float h[D_STATE];
#pragma unroll
    for (int n = 0; n < D_STATE; ++n) h[n] = 0.f;

    __shared__ float Bs[D_STATE];
    __shared__ float Cs[D_STATE];

    for (int l = 0; l < LL; ++l) {
        size_t row = (size_t)b * LL + l;
        if (threadIdx.x < 16)       Bs[threadIdx.x]      = dbl[row * 48 + 16 + threadIdx.x];
        else if (threadIdx.x < 32)  Cs[threadIdx.x - 16] = dbl[row * 48 + 32 + (threadIdx.x - 16)];
        __syncthreads();
        float dtv = dt[row * D_INNER + d];
        float u   = xc[row * D_INNER + d];
        float y = 0.f;
#pragma unroll
        for (int n = 0; n < D_STATE; ++n) {
            h[n] = h[n] * __expf(dtv * A[n]) + dtv * Bs[n] * u;
            y += h[n] * Cs[n];
        }
        ycore[row * D_INNER + d] = y + u * Dpv;
        __syncthreads();
    }
}

// ygated = ycore * silu(z), z = xz[:, 512:1024]; written over xc buffer
__global__ void gate_kernel(const float* __restrict__ ycore,
                            const float* __restrict__ xz,
                            float* __restrict__ yg)
{
    size_t idx = (size_t)blockIdx.x * blockDim.x + threadIdx.x;
    if (idx >= (size_t)ROWS * D_INNER) return;
    int d = idx % D_INNER;
    size_t row = idx / D_INNER;
    float z = xz[row * (2 * D_INNER) + D_INNER + d];
    yg[idx] = ycore[idx] * silu_f(z);
}

// ===========================================================================
// WMMA GEMM:  out[M,N] = act(A[M,K] @ W[N,K]^T + bias) (+ res)
// Each wave computes a 16x64 strip (4 N-subtiles): one A-fragment per k-step
// is reused across up to 4 B-fragments / 4 WMMAs. Accumulators are NAMED v8f
// variables (never dynamically indexed) so they stay in fixed VGPR octets.
// The k-loop is versioned on the (wave-uniform) subtile count so every hot
// loop is branch-free and loads clause-batch cleanly. EXEC is all-1s for
// every executed WMMA. act: 0 = none, 1 = softplus, 2 = exact GELU.
// ===========================================================================
__global__ void gemm_wmma_kernel(const float* __restrict__ A, int lda,
                                 const float* __restrict__ W,
                                 const float* __restrict__ bias,
                                 const float* __restrict__ res,
                                 float* __restrict__ out, int ldo,
                                 int M, int N, int K, int act)
{
    int wave = threadIdx.x >> 5;
    int lane = threadIdx.x & 31;
    int half = lane >> 4;          // 0: lanes 0-15, 1: lanes 16-31
    int lm   = lane & 15;
    int m0 = (blockIdx.x * 4 + wave) * 16;
    int n0 = blockIdx.y * 64;
    if (m0 >= M) return;
    int ntiles = (N - n0) / 16;            // wave-uniform
    if (ntiles > 4) ntiles = 4;

    const float* Ar  = A + (size_t)(m0 + lm) * lda;  // A-frag: lane = M row
    const float* Wr0 = W + (size_t)(n0 + lm) * K;    // B-frags: lane = N col
    const float* Wr1 = Wr0 + (size_t)16 * K;
    const float* Wr2 = Wr0 + (size_t)32 * K;
    const float* Wr3 = Wr0 + (size_t)48 * K;

    v8f acc0 = {}, acc1 = {}, acc2 = {}, acc3 = {};

    if (ntiles == 4) {
        for (int k0 = 0; k0 < K; k0 += 4) {
            int ka = k0 + 2 * half;
            v2f a, b0, b1, b2, b3;
            a.x  = Ar[ka];      a.y  = Ar[ka + 1];
            b0.x = Wr0[ka];     b0.y = Wr0[ka + 1];
            b1.x = Wr1[ka];     b1.y = Wr1[ka + 1];
            b2.x = Wr2[ka];     b2.y = Wr2[ka + 1];
            b3.x = Wr3[ka];     b3.y = Wr3[ka + 1];
            acc0 = __builtin_amdgcn_wmma_f32_16x16x4_f32(false, a, false, b0, (short)0, acc0, false, false);
            acc1 = __builtin_amdgcn_wmma_f32_16x16x4_f32(false, a, false, b1, (short)0, acc1, false, false);
            acc2 = __builtin_amdgcn_wmma_f32_16x16x4_f32(false, a, false, b2, (short)0, acc2, false, false);
            acc3 = __builtin_amdgcn_wmma_f32_16x16x4_f32(false, a, false, b3, (short)0, acc3, false, false);
        }
    } else if (ntiles == 3) {              // e.g. N = 48 (dbl GEMM)
        for (int k0 = 0; k0 < K; k0 += 4) {
            int ka = k0 + 2 * half;
            v2f a, b0, b1, b2;
            a.x  = Ar[ka];      a.y  = Ar[ka + 1];
            b0.x = Wr0[ka];     b0.y = Wr0[ka + 1];
            b1.x = Wr1[ka];     b1.y = Wr1[ka + 1];
            b2.x = Wr2[ka];     b2.y = Wr2[ka + 1];
            acc0 = __builtin_amdgcn_wmma_f32_16x16x4_f32(false, a, false, b0, (short)0, acc0, false, false);
            acc1 = __builtin_amdgcn_wmma_f32_16x16x4_f32(false, a, false, b1, (short)0, acc1, false, false);
            acc2 = __builtin_amdgcn_wmma_f32_16x16x4_f32(false, a, false, b2, (short)0, acc2, false, false);
        }
    } else if (ntiles == 2) {
        for (int k0 = 0; k0 < K; k0 += 4) {
            int ka = k0 + 2 * half;
            v2f a, b0, b1;
            a.x  = Ar[ka];      a.y  = Ar[ka + 1];
            b0.x = Wr0[ka];     b0.y = Wr0[ka + 1];
            b1.x = Wr1[ka];     b1.y = Wr1[ka + 1];
            acc0 = __builtin_amdgcn_wmma_f32_16x16x4_f32(false, a, false, b0, (short)0, acc0, false, false);
            acc1 = __builtin_amdgcn_wmma_f32_16x16x4_f32(false, a, false, b1, (short)0, acc1, false, false);
        }
    } else {
        for (int k0 = 0; k0 < K; k0 += 4) {
            int ka = k0 + 2 * half;
            v2f a, b0;
            a.x  = Ar[ka];      a.y  = Ar[ka + 1];
            b0.x = Wr0[ka];     b0.y = Wr0[ka + 1];
            acc0 = __builtin_amdgcn_wmma_f32_16x16x4_f32(false, a, false, b0, (short)0, acc0, false, false);
        }
    }

    // -------- epilogue: constant-index subtiles, uniform guards --------
    auto store_tile = [&](const v8f& c, int nt) {
        int col = n0 + nt * 16 + lm;
        float bv = bias ? bias[col] : 0.f;
#pragma unroll
        for (int i = 0; i < 8; ++i) {
            int row = m0 + i + 8 * half;
            float v = c[i] + bv;
            if (act == 1) {                  // softplus
                v = (v > 20.f) ? v : log1pf(__expf(v));
            } else if (act == 2) {           // exact GELU
                v = 0.5f * v * (1.0f + erff(v * 0.70710678118654752f));
            }
            size_t o = (size_t)row * ldo + col;
            if (res) v += res[o];
            out[o] = v;
        }
    };
    store_tile(acc0, 0);
    if (ntiles > 1) store_tile(acc1, 1);
    if (ntiles > 2) store_tile(acc2, 2);
    if (ntiles > 3) store_tile(acc3, 3);
}

// ===========================================================================
// Flash attention with fp32 WMMA. One wave per 16-row Q tile.
// qkv: (rows, 768) = [q | k | v], head h uses cols h*64 within each third.
// ===========================================================================
__global__ void attn_kernel(const float* __restrict__ qkv,
                            float* __restrict__ o)
{
    int wave = threadIdx.x >> 5;
    int lane = threadIdx.x & 31;
    int half = lane >> 4;
    int lm   = lane & 15;
    int b = blockIdx.x / N_HEADS;
    int h = blockIdx.x % N_HEADS;
    int q0 = (blockIdx.y * 4 + wave) * 16;

    __shared__ float plds[4][16 * 16];

    // Q fragments (16x64) held in registers: 16 v2f frags
    v2f aQ[16];
    {
        const float* qrow = qkv + ((size_t)(b * LL + q0 + lm)) * (3 * DD) + h * HD;
#pragma unroll
        for (int kk = 0; kk < 16; ++kk) {
            int ka = kk * 4 + 2 * half;
            aQ[kk].x = qrow[ka];
            aQ[kk].y = qrow[ka + 1];
        }
    }

    float Mrun[8], Lrun[8];
#pragma unroll
    for (int i = 0; i < 8; ++i) { Mrun[i] = -1e30f; Lrun[i] = 0.f; }
    v8f oc0 = {}, oc1 = {}, oc2 = {}, oc3 = {};

    const float scale = 0.125f;   // 1/sqrt(64)

    for (int j = 0; j < LL / 16; ++j) {
        // ---- S = Q K^T (16x16) ----
        v8f s = {};
        const float* krow = qkv + ((size_t)(b * LL + j * 16 + lm)) * (3 * DD) + DD + h * HD;
#pragma unroll
        for (int kk = 0; kk < 16; ++kk) {
            int ka = kk * 4 + 2 * half;
            v2f bK;
            bK.x = krow[ka];
            bK.y = krow[ka + 1];
            s = __builtin_amdgcn_wmma_f32_16x16x4_f32(false, aQ[kk], false, bK,
                                                      (short)0, s, false, false);
        }

        // ---- online softmax over each 16-lane row segment ----
        float p[8];
#pragma unroll
        for (int i = 0; i < 8; ++i) {
            float sv = s[i] * scale;
            float t = sv;
#pragma unroll
            for (int off = 8; off >= 1; off >>= 1) t = fmaxf(t, __shfl_xor(t, off, 16));
            float mn = fmaxf(Mrun[i], t);
            float al = __expf(Mrun[i] - mn);
            float pv = __expf(sv - mn);
            float r = pv;
#pragma unroll
            for (int off = 8; off >= 1; off >>= 1) r += __shfl_xor(r, off, 16);
            Lrun[i] = Lrun[i] * al + r;
            Mrun[i] = mn;
            p[i] = pv;
            oc0[i] *= al;
            oc1[i] *= al;
            oc2[i] *= al;
            oc3[i] *= al;
        }

        // ---- C-layout -> A-layout for P via LDS ----
        __syncthreads();
#pragma unroll
        for (int i = 0; i < 8; ++i)
            plds[wave][(i + 8 * half) * 16 + lm] = p[i];
        __syncthreads();

        v2f aP[4];
#pragma unroll
        for (int kt = 0; kt < 4; ++kt) {
            int ka = kt * 4 + 2 * half;
            aP[kt].x = plds[wave][lm * 16 + ka];
            aP[kt].y = plds[wave][lm * 16 + ka + 1];
        }

        // ---- O += P V  (16x16 x 16x64) ----
        const float* vbase = qkv + ((size_t)(b * LL + j * 16)) * (3 * DD) + 2 * DD + h * HD;
#pragma unroll
        for (int kt = 0; kt < 4; ++kt) {
            int kidx = kt * 4 + 2 * half;
            v2f bV0, bV1, bV2, bV3;
            bV0.x = vbase[(size_t)kidx * (3 * DD) +  0 + lm];
            bV0.y = vbase[(size_t)(kidx + 1) * (3 * DD) +  0 + lm];
            bV1.x = vbase[(size_t)kidx * (3 * DD) + 16 + lm];
            bV1.y = vbase[(size_t)(kidx + 1) * (3 * DD) + 16 + lm];
            bV2.x = vbase[(size_t)kidx * (3 * DD) + 32 + lm];
            bV2.y = vbase[(size_t)(kidx + 1) * (3 * DD) + 32 + lm];
            bV3.x = vbase[(size_t)kidx * (3 * DD) + 48 + lm];
            bV3.y = vbase[(size_t)(kidx + 1) * (3 * DD) + 48 + lm];
            oc0 = __builtin_amdgcn_wmma_f32_16x16x4_f32(false, aP[kt], false, bV0, (short)0, oc0, false, false);
            oc1 = __builtin_amdgcn_wmma_f32_16x16x4_f32(false, aP[kt], false, bV1, (short)0, oc1, false, false);
            oc2 = __builtin_amdgcn_wmma_f32_16x16x4_f32(false, aP[kt], false, bV2, (short)0, oc2, false, false);
            oc3 = __builtin_amdgcn_wmma_f32_16x16x4_f32(false, aP[kt], false, bV3, (short)0, oc3, false, false);
        }
    }

    // ---- epilog: normalize and store into (rows, 256) ----
    auto store_o = [&](const v8f& c, int t4) {
#pragma unroll
        for (int i = 0; i < 8; ++i) {
            int row = q0 + i + 8 * half;
            int col = h * HD + t4 * 16 + lm;
            o[((size_t)(b * LL + row)) * DD + col] = c[i] / Lrun[i];
        }
    };
    store_o(oc0, 0);
    store_o(oc1, 1);
    store_o(oc2, 2);
    store_o(oc3, 3);
}

// ===========================================================================
// Host side
// ===========================================================================
extern "C" void kernel_launch(void* const* d_in, const int* in_sizes, int n_in,
                              void* d_out, int out_size, void* d_ws, size_t ws_size,
                              hipStream_t stream)
{
    (void)in_sizes; (void)n_in; (void)out_size; (void)ws_size;
    const float* x     = (const float*)d_in[0];
    // mf params: 1..9 ; mb params: 10..18
    const float* n1w = (const float*)d_in[19];
    const float* n1b = (const float*)d_in[20];
    const float* n2w = (const float*)d_in[21];
    const float* n2b = (const float*)d_in[22];
    const float* n3w = (const float*)d_in[23];
    const float* n3b = (const float*)d_in[24];
    const float* Wqkv = (const float*)d_in[25];
    const float* bqkv = (const float*)d_in[26];
    const float* Wo   = (const float*)d_in[27];
    const float* bo   = (const float*)d_in[28];
    const float* fc1w = (const float*)d_in[29];
    const float* fc1b = (const float*)d_in[30];
    const float* fc2w = (const float*)d_in[31];
    const float* fc2b = (const float*)d_in[32];

    // -------- workspace layout (floats) --------
    float* ws = (float*)d_ws;
    size_t off = 0;
    auto alloc = [&](size_t n) { float* p = ws + off; off += n; return p; };
    float* h      = alloc((size_t)ROWS * DD);          // LN1 out / LN2 out (reused)
    float* hflip  = alloc((size_t)ROWS * DD);          // flipped h / LN3 out (reused)
    float* xz     = alloc((size_t)ROWS * 2 * D_INNER); // xz  / qkv / mlp-mid (reused)
    float* xc     = alloc((size_t)ROWS * D_INNER);     // conv out -> ygated / attn_o (reused)
    float* dblb   = alloc((size_t)ROWS * 48);
    float* dtb    = alloc((size_t)ROWS * D_INNER);
    float* ycore  = alloc((size_t)ROWS * D_INNER);
    float* out_f  = alloc((size_t)ROWS * DD);
    float* out_b  = alloc((size_t)ROWS * DD);
    float* x1     = alloc((size_t)ROWS * DD);
    float* x2     = alloc((size_t)ROWS * DD);

    const int TPB = 256;
    size_t nXD  = (size_t)ROWS * DD;        // 2M
    size_t nXI  = (size_t)ROWS * D_INNER;   // 4M
    dim3 gemmBlk(128);

    auto gemm = [&](const float* A, int lda, const float* W, const float* bias,
                    const float* res, float* out, int ldo, int N, int K, int act) {
        dim3 grid(ROWS / 64, (N + 63) / 64);
        gemm_wmma_kernel<<<grid, gemmBlk, 0, stream>>>(A, lda, W, bias, res,
                                                       out, ldo, ROWS, N, K, act);
    };

    auto run_mamba = [&](const float* hin, int base, float* outbuf) {
        const float* Win   = (const float*)d_in[base + 0];
        const float* convw = (const float*)d_in[base + 1];
        const float* convb = (const float*)d_in[base + 2];
        const float* Wx    = (const float*)d_in[base + 3];
        const float* Wdt   = (const float*)d_in[base + 4];
        const float* bdt   = (const float*)d_in[base + 5];
        const float* Alog  = (const float*)d_in[base + 6];
        const float* Dp    = (const float*)d_in[base + 7];
        const float* Wout  = (const float*)d_in[base + 8];

        // xz = h @ Win^T                    (8192 x 1024, K=256)
        gemm(hin, DD, Win, nullptr, nullptr, xz, 2 * D_INNER, 2 * D_INNER, DD, 0);
        // depthwise conv + bias + SiLU -> xc
        conv_silu_kernel<<<(nXI + TPB - 1) / TPB, TPB, 0, stream>>>(xz, convw, convb, xc);
        // dbl = xc @ Wx^T                   (8192 x 48,   K=512)
        gemm(xc, D_INNER, Wx, nullptr, nullptr, dblb, 48, 48, D_INNER, 0);
        // dt = softplus(dbl[:, :16] @ Wdt^T + bdt)   (8192 x 512, K=16)
        gemm(dblb, 48, Wdt, bdt, nullptr, dtb, D_INNER, D_INNER, DT_RANK, 1);
        // selective scan
        scan_kernel<<<dim3(BB, D_INNER / 256), 256, 0, stream>>>(dtb, dblb, xc,
                                                                 Alog, Dp, ycore);
        // ygated = ycore * silu(z)  (overwrites xc)
        gate_kernel<<<(nXI + TPB - 1) / TPB, TPB, 0, stream>>>(ycore, xz, xc);
        // out = ygated @ Wout^T             (8192 x 256,  K=512)
        gemm(xc, D_INNER, Wout, nullptr, nullptr, outbuf, DD, DD, D_INNER, 0);
    };

    // ---- stage 1: LN1 + bidirectional Mamba ----
    ln_kernel<<<ROWS / 4, 128, 0, stream>>>(x, n1w, n1b, h, ROWS);
    flip_kernel<<<(nXD + TPB - 1) / TPB, TPB, 0, stream>>>(h, hflip);
    run_mamba(h,     1,  out_f);
    run_mamba(hflip, 10, out_b);
    combine_kernel<<<(nXD + TPB - 1) / TPB, TPB, 0, stream>>>(x, out_f, out_b, x1);

    // ---- stage 2: attention ----
    ln_kernel<<<ROWS / 4, 128, 0, stream>>>(x1, n2w, n2b, h, ROWS);
    gemm(h, DD, Wqkv, bqkv, nullptr, xz, 3 * DD, 3 * DD, DD, 0);          // qkv
    attn_kernel<<<dim3(BB * N_HEADS, LL / 64), 128, 0, stream>>>(xz, xc); // attn_o
    gemm(xc, DD, Wo, bo, x1, x2, DD, DD, DD, 0);                          // + residual

    // ---- stage 3: MLP ----
    ln_kernel<<<ROWS / 4, 128, 0, stream>>>(x2, n3w, n3b, hflip, ROWS);
    gemm(hflip, DD, fc1w, fc1b, nullptr, xz, H_MLP, H_MLP, DD, 2);        // GELU
    gemm(xz, H_MLP, fc2w, fc2b, x2, (float*)d_out, DD, DD, H_MLP, 0);     // + residual
}